// BatchAllTripletLoss_2293512536517
// MI455X (gfx1250) — compile-verified
//
#include <hip/hip_runtime.h>
#include <hip/hip_bf16.h>

// ---------------------------------------------------------------------------
// BatchAllTripletLoss for MI455X (gfx1250, wave32, WMMA)
//   E: [512, 256] fp32 embeddings, labels: [512] int32 -> scalar fp32 loss
// Pipeline (all on `stream`):
//   1) init_accum      : zero the (sum, count) accumulators in ws
//   2) row_sqnorm      : sq[i] = ||E_i||^2            (one wave per row)
//   3) gram_dist_wmma  : dist[i][j] = sqrt(max(sq_i+sq_j-2*(E E^T)_ij, 0))
//                        via V_WMMA_F32_16X16X4_F32, fused epilogue
//   4) triplet_reduce  : per-anchor LDS-resident (p,n) loop, relu-sum + count
//   5) finalize        : out = sum / count
// ---------------------------------------------------------------------------

#define TN_N 512
#define TN_D 256
#define TN_MARGIN 1.0f

typedef __attribute__((ext_vector_type(2))) float v2f;
typedef __attribute__((ext_vector_type(8))) float v8f;

__global__ void tl_init_accum(float* __restrict__ accum) {
  if (threadIdx.x < 2) accum[threadIdx.x] = 0.0f;
}

// One wave (32 lanes) per row: each lane sums 8 elements, then shfl-xor tree.
__global__ void tl_row_sqnorm(const float* __restrict__ e,
                              float* __restrict__ sq) {
  const int row  = blockIdx.x;
  const int lane = threadIdx.x & 31;
  const float* r = e + row * TN_D;
  float s = 0.0f;
#pragma unroll
  for (int k = lane; k < TN_D; k += 32) {
    float v = r[k];
    s = fmaf(v, v, s);
  }
#pragma unroll
  for (int off = 16; off > 0; off >>= 1) s += __shfl_xor(s, off, 32);
  if (lane == 0) sq[row] = s;
}

// Gram + distance, one 16x16 output tile per wave.
// FP32 WMMA ISA layouts (CDNA5 ISA):
//   A 16x4 : lanes 0-15 row M=lane,  VGPR{0,1} = K={0,1}; lanes 16-31 K={2,3}
//   B 4x16 : mirrored with N in place of M (B = E^T, so B column n is E row tn+n)
//   C/D    : VGPR v -> M = v + 8*(lane>=16), N = lane&15
__global__ void tl_gram_dist_wmma(const float* __restrict__ e,
                                  const float* __restrict__ sq,
                                  float* __restrict__ dist) {
  const int wavesPerBlock = blockDim.x >> 5;
  const int wave = blockIdx.x * wavesPerBlock + (threadIdx.x >> 5);
  const int lane = threadIdx.x & 31;
  const int tilesPerRow = TN_N / 16;            // 32
  const int tm = (wave / tilesPerRow) * 16;     // tile row base
  const int tn = (wave % tilesPerRow) * 16;     // tile col base
  const int half = lane >> 4;                   // 0: K pair {k,k+1}, 1: {k+2,k+3}
  const int lm   = lane & 15;

  const float* arow = e + (tm + lm) * TN_D;     // A row (M = tm+lm)
  const float* brow = e + (tn + lm) * TN_D;     // B "column" (N = tn+lm) of E^T

  v8f c = {};
#pragma unroll 4
  for (int k = 0; k < TN_D; k += 4) {
    // 8-byte aligned vector loads of the lane's K-pair
    v2f a = *(const v2f*)(arow + k + 2 * half);
    v2f b = *(const v2f*)(brow + k + 2 * half);
    // D = A x B + C : v_wmma_f32_16x16x4_f32
    c = __builtin_amdgcn_wmma_f32_16x16x4_f32(
        /*neg_a=*/false, a, /*neg_b=*/false, b,
        /*c_mod=*/(short)0, c, /*reuse_a=*/false, /*reuse_b=*/false);
  }

  // Fused distance epilogue: d2 = sq_i + sq_j - 2*gram, safe sqrt.
  const float sqj = sq[tn + lm];
#pragma unroll
  for (int v = 0; v < 8; ++v) {
    const int i  = tm + v + 8 * half;
    float d2 = sq[i] + sqj - 2.0f * c[v];
    d2 = fmaxf(d2, 0.0f);
    dist[i * TN_N + (tn + lm)] = __builtin_sqrtf(d2);
  }
}

// One block per anchor. Distance row + labels cached in LDS.
// loss contribution: sum_{p in P(a)} sum_{n in Neg(a)} relu(d_ap - d_an + m)
// count contribution: |P(a)| * |Neg(a)|
__global__ void tl_triplet_reduce(const float* __restrict__ dist,
                                  const int* __restrict__ labels,
                                  float* __restrict__ accum) {
  __shared__ float drow[TN_N];
  __shared__ int   lsh[TN_N];
  __shared__ float redS[8];
  __shared__ float redC[8];

  const int a   = blockIdx.x;
  const int tid = threadIdx.x;

  for (int j = tid; j < TN_N; j += blockDim.x) {
    drow[j] = dist[a * TN_N + j];
    lsh[j]  = labels[j];
  }
  __syncthreads();

  const int la = lsh[a];
  float s = 0.0f, cnt = 0.0f;

  for (int p = 0; p < TN_N; ++p) {
    if (p == a || lsh[p] != la) continue;      // need a valid positive
    const float thresh = drow[p] + TN_MARGIN;  // d_ap + margin
    for (int n = tid; n < TN_N; n += blockDim.x) {
      if (lsh[n] == la) continue;              // need a valid negative
      const float t = thresh - drow[n];
      s += (t > 0.0f) ? t : 0.0f;
      cnt += 1.0f;
    }
  }

  // wave32 tree reduction, then cross-wave via LDS
#pragma unroll
  for (int off = 16; off > 0; off >>= 1) {
    s   += __shfl_xor(s, off, 32);
    cnt += __shfl_xor(cnt, off, 32);
  }
  const int w = tid >> 5, lane = tid & 31;
  if (lane == 0) { redS[w] = s; redC[w] = cnt; }
  __syncthreads();
  if (tid == 0) {
    float S = 0.0f, C = 0.0f;
    const int nw = blockDim.x >> 5;
    for (int i = 0; i < nw; ++i) { S += redS[i]; C += redC[i]; }
    atomicAdd(&accum[0], S);
    atomicAdd(&accum[1], C);
  }
}

__global__ void tl_finalize(const float* __restrict__ accum,
                            float* __restrict__ out) {
  if (threadIdx.x == 0) out[0] = accum[0] / accum[1];
}

extern "C" void kernel_launch(void* const* d_in, const int* in_sizes, int n_in,
                              void* d_out, int out_size, void* d_ws, size_t ws_size,
                              hipStream_t stream) {
  (void)in_sizes; (void)n_in; (void)out_size; (void)ws_size;
  const float* e      = (const float*)d_in[0];   // [512, 256] fp32
  const int*   labels = (const int*)d_in[1];     // [512] int32
  float*       out    = (float*)d_out;           // scalar fp32

  // Workspace layout (floats): [0,512) sq | [512,514) accum | [1024, 1024+512*512) dist
  float* ws    = (float*)d_ws;
  float* sq    = ws;
  float* accum = ws + 512;
  float* dist  = ws + 1024;

  tl_init_accum<<<1, 32, 0, stream>>>(accum);
  tl_row_sqnorm<<<TN_N, 32, 0, stream>>>(e, sq);

  // 1024 tiles of 16x16, 8 waves (256 threads) per block -> 128 blocks
  tl_gram_dist_wmma<<<(TN_N / 16) * (TN_N / 16) / 8, 256, 0, stream>>>(e, sq, dist);

  tl_triplet_reduce<<<TN_N, 256, 0, stream>>>(dist, labels, accum);
  tl_finalize<<<1, 32, 0, stream>>>(accum, out);
}